// MedianPool2d_1236950581312
// MI455X (gfx1250) — compile-verified
//
#include <hip/hip_runtime.h>

// 3x3 median filter, stride 1, reflect pad 1, on [8,16,512,512] fp32.
// Memory-bound: float4 (b128) loads/stores, exact median-of-9 sorting network.

#define IMG_H 512
#define IMG_W 512

__device__ __forceinline__ void sort2(float& a, float& b) {
    float lo = fminf(a, b);
    b = fmaxf(a, b);
    a = lo;
}

// Exact median of 9 (Devillard / Smith 19-exchange network).
__device__ __forceinline__ float median9(float p0, float p1, float p2,
                                         float p3, float p4, float p5,
                                         float p6, float p7, float p8) {
    sort2(p1, p2); sort2(p4, p5); sort2(p7, p8);
    sort2(p0, p1); sort2(p3, p4); sort2(p6, p7);
    sort2(p1, p2); sort2(p4, p5); sort2(p7, p8);
    sort2(p0, p3); sort2(p5, p8); sort2(p4, p7);
    sort2(p3, p6); sort2(p1, p4); sort2(p2, p5);
    sort2(p4, p7); sort2(p4, p2); sort2(p6, p4);
    sort2(p4, p2);
    return p4;
}

__global__ __launch_bounds__(256) void median3x3_kernel(
    const float* __restrict__ in, float* __restrict__ out, int n_chunks) {
    const int chunk = blockIdx.x * 256 + threadIdx.x;
    if (chunk >= n_chunks) return;

    // 128 float4-chunks per row, 512 rows per image, B*C images.
    const int c0  = (chunk & 127) << 2;        // starting column (0,4,...,508)
    const int r   = (chunk >> 7) & (IMG_H - 1);
    const int img = chunk >> 16;               // 128*512 = 2^16 chunks per image

    const float* base  = in  + (size_t)img * (IMG_H * IMG_W);
    float*       obase = out + (size_t)img * (IMG_H * IMG_W);

    // Reflect padding (pad=1): -1 -> 1, 512 -> 510.
    const int rm   = (r == 0)          ? 1            : r - 1;
    const int rp   = (r == IMG_H - 1)  ? IMG_H - 2    : r + 1;
    const int lcol = (c0 == 0)         ? 1            : c0 - 1;
    const int rcol = (c0 == IMG_W - 4) ? IMG_W - 2    : c0 + 4;

    const int rr[3] = {rm, r, rp};
    float w[3][6];
#pragma unroll
    for (int k = 0; k < 3; ++k) {
        const float* rowp = base + rr[k] * IMG_W;
        float4 v = *reinterpret_cast<const float4*>(rowp + c0);  // global_load_b128
        w[k][0] = rowp[lcol];
        w[k][1] = v.x; w[k][2] = v.y; w[k][3] = v.z; w[k][4] = v.w;
        w[k][5] = rowp[rcol];
    }

    float4 o;
    o.x = median9(w[0][0], w[0][1], w[0][2],
                  w[1][0], w[1][1], w[1][2],
                  w[2][0], w[2][1], w[2][2]);
    o.y = median9(w[0][1], w[0][2], w[0][3],
                  w[1][1], w[1][2], w[1][3],
                  w[2][1], w[2][2], w[2][3]);
    o.z = median9(w[0][2], w[0][3], w[0][4],
                  w[1][2], w[1][3], w[1][4],
                  w[2][2], w[2][3], w[2][4]);
    o.w = median9(w[0][3], w[0][4], w[0][5],
                  w[1][3], w[1][4], w[1][5],
                  w[2][3], w[2][4], w[2][5]);

    *reinterpret_cast<float4*>(obase + r * IMG_W + c0) = o;  // global_store_b128
}

extern "C" void kernel_launch(void* const* d_in, const int* in_sizes, int n_in,
                              void* d_out, int out_size, void* d_ws, size_t ws_size,
                              hipStream_t stream) {
    (void)in_sizes; (void)n_in; (void)d_ws; (void)ws_size;
    const float* x = (const float*)d_in[0];
    float* y = (float*)d_out;

    const int n_chunks = out_size >> 2;               // 4 outputs per thread
    const int blocks = (n_chunks + 255) / 256;        // 32768 for 8*16*512*512
    median3x3_kernel<<<blocks, 256, 0, stream>>>(x, y, n_chunks);
}